// Gamba_6030134083940
// MI455X (gfx1250) — compile-verified
//
#include <hip/hip_runtime.h>
#include <hip/hip_bf16.h>
#include <math.h>

// ---------------- problem constants ----------------
#define NTOT   32768
#define BSZ    64
#define NNODE  512
#define EDGES  262144
#define HID    128
#define INC    128
#define OUTC   32
#define LPE_N  8
#define RWL_N  16
#define DM     152     // HID + LPE + RWL
#define KTOK   4
#define DSTATE 128
#define DCONV  4
#define DINNER 304
#define DTRANK 10
#define MERGEK 280     // HID + DM

typedef float v2f __attribute__((ext_vector_type(2)));
typedef float v8f __attribute__((ext_vector_type(8)));

// padded LDS strides (avoid 16-way bank conflicts on A-fragment column reads)
#define XSTRIDE  132
#define MSTRIDE  284

// =====================================================================
// Fused MLP2: Y = relu(LN(X@W1 + b1)) @ W2 + b2     (K = 128 both GEMMs)
// X_eff = scale * xa + xadd.  128 threads = 4 waves, 64 rows / WG.
// Uses V_WMMA_F32_16X16X4_F32.
// =====================================================================
template <int NOUT2>
__global__ __launch_bounds__(128) void mlp2_wmma_kernel(
    const float* __restrict__ xa, const float* __restrict__ xadd,
    const float* __restrict__ epsp,
    const float* __restrict__ W1, const float* __restrict__ b1,
    const float* __restrict__ gam, const float* __restrict__ bet,
    const float* __restrict__ W2, const float* __restrict__ b2,
    float* __restrict__ out)
{
    __shared__ float tile[64 * XSTRIDE];

    const int tid  = threadIdx.x;
    const int wave = tid >> 5;
    const int lane = tid & 31;
    const int lr   = lane & 15;
    const int lh   = lane >> 4;
    const int rowbase = blockIdx.x * 64;

    float scale = 1.0f;
    if (epsp) scale = 1.0f + epsp[0];

    // load 64x128 X tile
    for (int i = tid; i < 64 * 128; i += 128) {
        int r = i >> 7, c = i & 127;
        int gi = (rowbase + r) * 128 + c;
        float v = xa[gi] * scale;
        if (xadd) v += xadd[gi];
        tile[r * XSTRIDE + c] = v;
    }
    __syncthreads();

    const int mrow = wave * 16 + lr;

    // ---- GEMM1: T = X @ W1 (128x128) ----
    v8f acc[8];
    for (int nt = 0; nt < 8; ++nt)
        for (int r = 0; r < 8; ++r) acc[nt][r] = 0.0f;

    for (int kk = 0; kk < 32; ++kk) {
        const int kb = kk * 4;
        v2f a;
        a[0] = tile[mrow * XSTRIDE + kb + 2 * lh + 0];
        a[1] = tile[mrow * XSTRIDE + kb + 2 * lh + 1];
#pragma unroll
        for (int nt = 0; nt < 8; ++nt) {
            v2f bf;
            bf[0] = W1[(kb + 2 * lh + 0) * 128 + nt * 16 + lr];
            bf[1] = W1[(kb + 2 * lh + 1) * 128 + nt * 16 + lr];
            acc[nt] = __builtin_amdgcn_wmma_f32_16x16x4_f32(
                false, a, false, bf, (short)0, acc[nt], false, false);
        }
    }
    __syncthreads();   // all X reads done -> reuse tile for T

    // store T = acc + b1 into LDS (C layout: reg r -> M = r + 8*lh)
#pragma unroll
    for (int nt = 0; nt < 8; ++nt)
#pragma unroll
        for (int r = 0; r < 8; ++r) {
            int m = wave * 16 + r + 8 * lh;
            int n = nt * 16 + lr;
            tile[m * XSTRIDE + n] = acc[nt][r] + b1[n];
        }
    __syncthreads();

    // ---- LayerNorm + ReLU, one row per thread (rows 0..63) ----
    if (tid < 64) {
        float* row = &tile[tid * XSTRIDE];
        float mu = 0.0f;
        for (int c = 0; c < 128; ++c) mu += row[c];
        mu *= (1.0f / 128.0f);
        float var = 0.0f;
        for (int c = 0; c < 128; ++c) { float d = row[c] - mu; var += d * d; }
        var *= (1.0f / 128.0f);
        float rs = rsqrtf(var + 1e-5f);
        for (int c = 0; c < 128; ++c) {
            float v = (row[c] - mu) * rs * gam[c] + bet[c];
            row[c] = fmaxf(v, 0.0f);
        }
    }
    __syncthreads();

    // ---- GEMM2: Y = T @ W2 (128 x NOUT2) ----
    constexpr int NT2 = NOUT2 / 16;
    v8f acc2[NT2];
    for (int nt = 0; nt < NT2; ++nt)
        for (int r = 0; r < 8; ++r) acc2[nt][r] = 0.0f;

    for (int kk = 0; kk < 32; ++kk) {
        const int kb = kk * 4;
        v2f a;
        a[0] = tile[mrow * XSTRIDE + kb + 2 * lh + 0];
        a[1] = tile[mrow * XSTRIDE + kb + 2 * lh + 1];
#pragma unroll
        for (int nt = 0; nt < NT2; ++nt) {
            v2f bf;
            bf[0] = W2[(kb + 2 * lh + 0) * NOUT2 + nt * 16 + lr];
            bf[1] = W2[(kb + 2 * lh + 1) * NOUT2 + nt * 16 + lr];
            acc2[nt] = __builtin_amdgcn_wmma_f32_16x16x4_f32(
                false, a, false, bf, (short)0, acc2[nt], false, false);
        }
    }
#pragma unroll
    for (int nt = 0; nt < NT2; ++nt)
#pragma unroll
        for (int r = 0; r < 8; ++r) {
            int m = rowbase + wave * 16 + r + 8 * lh;
            int n = nt * 16 + lr;
            out[m * NOUT2 + n] = acc2[nt][r] + b2[n];
        }
}

// =====================================================================
// Single GEMM (K=128, N=128): Y = (xa + xadd) @ W + b   (output GINConv)
// =====================================================================
__global__ __launch_bounds__(128) void gemm128_wmma_kernel(
    const float* __restrict__ xa, const float* __restrict__ xadd,
    const float* __restrict__ W, const float* __restrict__ bias,
    float* __restrict__ out)
{
    __shared__ float tile[64 * XSTRIDE];
    const int tid  = threadIdx.x;
    const int wave = tid >> 5;
    const int lane = tid & 31;
    const int lr = lane & 15, lh = lane >> 4;
    const int rowbase = blockIdx.x * 64;

    for (int i = tid; i < 64 * 128; i += 128) {
        int r = i >> 7, c = i & 127;
        int gi = (rowbase + r) * 128 + c;
        tile[r * XSTRIDE + c] = xa[gi] + xadd[gi];
    }
    __syncthreads();

    const int mrow = wave * 16 + lr;
    v8f acc[8];
    for (int nt = 0; nt < 8; ++nt)
        for (int r = 0; r < 8; ++r) acc[nt][r] = 0.0f;

    for (int kk = 0; kk < 32; ++kk) {
        const int kb = kk * 4;
        v2f a;
        a[0] = tile[mrow * XSTRIDE + kb + 2 * lh + 0];
        a[1] = tile[mrow * XSTRIDE + kb + 2 * lh + 1];
#pragma unroll
        for (int nt = 0; nt < 8; ++nt) {
            v2f bf;
            bf[0] = W[(kb + 2 * lh + 0) * 128 + nt * 16 + lr];
            bf[1] = W[(kb + 2 * lh + 1) * 128 + nt * 16 + lr];
            acc[nt] = __builtin_amdgcn_wmma_f32_16x16x4_f32(
                false, a, false, bf, (short)0, acc[nt], false, false);
        }
    }
#pragma unroll
    for (int nt = 0; nt < 8; ++nt)
#pragma unroll
        for (int r = 0; r < 8; ++r) {
            int m = rowbase + wave * 16 + r + 8 * lh;
            int n = nt * 16 + lr;
            out[m * 128 + n] = acc[nt][r] + bias[n];
        }
}

// =====================================================================
// Merge GEMM: out = [h2 | last[batch]] @ W_merge + b   (K=280, N=128)
// 64 threads = 2 waves, 32 rows / WG.
// =====================================================================
__global__ __launch_bounds__(64) void merge_wmma_kernel(
    const float* __restrict__ h2, const float* __restrict__ lastv,
    const int* __restrict__ batch,
    const float* __restrict__ Wm, const float* __restrict__ bm,
    float* __restrict__ out)
{
    __shared__ float tile[32 * MSTRIDE];
    const int tid  = threadIdx.x;
    const int wave = tid >> 5;
    const int lane = tid & 31;
    const int lr = lane & 15, lh = lane >> 4;
    const int rowbase = blockIdx.x * 32;

    for (int i = tid; i < 32 * MERGEK; i += 64) {
        int r = i / MERGEK, c = i % MERGEK;
        int gr = rowbase + r;
        float v = (c < 128) ? h2[gr * 128 + c]
                            : lastv[batch[gr] * DM + (c - 128)];
        tile[r * MSTRIDE + c] = v;
    }
    __syncthreads();

    const int mrow = wave * 16 + lr;
    v8f acc[8];
    for (int nt = 0; nt < 8; ++nt)
        for (int r = 0; r < 8; ++r) acc[nt][r] = 0.0f;

    for (int kk = 0; kk < 70; ++kk) {   // 280 / 4
        const int kb = kk * 4;
        v2f a;
        a[0] = tile[mrow * MSTRIDE + kb + 2 * lh + 0];
        a[1] = tile[mrow * MSTRIDE + kb + 2 * lh + 1];
#pragma unroll
        for (int nt = 0; nt < 8; ++nt) {
            v2f bf;
            bf[0] = Wm[(kb + 2 * lh + 0) * 128 + nt * 16 + lr];
            bf[1] = Wm[(kb + 2 * lh + 1) * 128 + nt * 16 + lr];
            acc[nt] = __builtin_amdgcn_wmma_f32_16x16x4_f32(
                false, a, false, bf, (short)0, acc[nt], false, false);
        }
    }
#pragma unroll
    for (int nt = 0; nt < 8; ++nt)
#pragma unroll
        for (int r = 0; r < 8; ++r) {
            int m = rowbase + wave * 16 + r + 8 * lh;
            int n = nt * 16 + lr;
            out[m * 128 + n] = acc[nt][r] + bm[n];
        }
}

// =====================================================================
// helpers / small kernels
// =====================================================================
__global__ void zero_kernel(float* __restrict__ p, int n)
{
    int i = blockIdx.x * blockDim.x + threadIdx.x;
    if (i < n) p[i] = 0.0f;
}

// scatter-add: agg[dst] += h[src], 4 floats per thread
__global__ void segsum_kernel(const float* __restrict__ h,
                              const int* __restrict__ src,
                              const int* __restrict__ dst,
                              float* __restrict__ agg)
{
    int i = blockIdx.x * blockDim.x + threadIdx.x;   // over EDGES*32
    int e  = i >> 5;
    int c4 = (i & 31) << 2;
    int s = src[e], d = dst[e];
    const float4 v = *(const float4*)&h[s * 128 + c4];
    float* o = &agg[d * 128 + c4];
    atomicAdd(o + 0, v.x);
    atomicAdd(o + 1, v.y);
    atomicAdd(o + 2, v.z);
    atomicAdd(o + 3, v.w);
}

__device__ __forceinline__ float hp_elem(const float* h2, const float* lpe,
                                         const float* rwse, int node, int c)
{
    if (c < 128) return h2[node * 128 + c];
    if (c < 128 + LPE_N) return lpe[node * LPE_N + (c - 128)];
    return rwse[node * RWL_N + (c - 128 - LPE_N)];
}

// alpha[i, 0..3] = hp(i,:) @ W_theta
__global__ void alpha_kernel(const float* __restrict__ h2,
                             const float* __restrict__ lpe,
                             const float* __restrict__ rwse,
                             const float* __restrict__ Wt,
                             float* __restrict__ alpha)
{
    int i = blockIdx.x * blockDim.x + threadIdx.x;
    if (i >= NTOT) return;
    float a0 = 0.f, a1 = 0.f, a2 = 0.f, a3 = 0.f;
    for (int c = 0; c < DM; ++c) {
        float v = hp_elem(h2, lpe, rwse, i, c);
        const float* w = &Wt[c * 4];
        a0 += v * w[0]; a1 += v * w[1]; a2 += v * w[2]; a3 += v * w[3];
    }
    float* o = &alpha[i * 4];
    o[0] = a0; o[1] = a1; o[2] = a2; o[3] = a3;
}

// per-graph: argmax over nodes per slot (first-max ties), gather hp/N
__global__ __launch_bounds__(256) void tokens_kernel(
    const float* __restrict__ alpha,
    const float* __restrict__ h2, const float* __restrict__ lpe,
    const float* __restrict__ rwse, float* __restrict__ tokens)
{
    const int b = blockIdx.x, tid = threadIdx.x;
    __shared__ float sv[256];
    __shared__ int   si[256];

    for (int k = 0; k < KTOK; ++k) {
        float best = -INFINITY;
        int bi = 0x7fffffff;
        for (int n = tid; n < NNODE; n += 256) {
            float v = alpha[(b * NNODE + n) * 4 + k];
            if (v > best || (v == best && n < bi)) { best = v; bi = n; }
        }
        sv[tid] = best; si[tid] = bi;
        __syncthreads();
        for (int off = 128; off > 0; off >>= 1) {
            if (tid < off) {
                float v2 = sv[tid + off]; int i2 = si[tid + off];
                if (v2 > sv[tid] || (v2 == sv[tid] && i2 < si[tid])) {
                    sv[tid] = v2; si[tid] = i2;
                }
            }
            __syncthreads();
        }
        int idx = si[0];
        int node = b * NNODE + idx;
        for (int c = tid; c < DM; c += 256)
            tokens[(b * KTOK + k) * DM + c] =
                hp_elem(h2, lpe, rwse, node, c) * (1.0f / (float)NNODE);
        __syncthreads();
    }
}

// =====================================================================
// Mamba block for one batch element; only last-token output is needed.
// =====================================================================
__global__ __launch_bounds__(256) void mamba_kernel(
    const float* __restrict__ tokens,
    const float* __restrict__ W_in,   const float* __restrict__ conv_w,
    const float* __restrict__ conv_b, const float* __restrict__ W_xproj,
    const float* __restrict__ W_dtproj, const float* __restrict__ b_dtproj,
    const float* __restrict__ A_log, const float* __restrict__ D_m,
    const float* __restrict__ W_outp,
    const float* __restrict__ g_lnm, const float* __restrict__ b_lnm,
    float* __restrict__ lastv)
{
    const int b = blockIdx.x, tid = threadIdx.x;
    __shared__ float u[KTOK][DM];
    __shared__ float xz[KTOK][2 * DINNER];
    __shared__ float xc[KTOK][DINNER];
    __shared__ float proj[KTOK][DTRANK + 2 * DSTATE];
    __shared__ float dtv[KTOK][DINNER];
    __shared__ float y3[DINNER];
    __shared__ float val[DINNER];
    __shared__ float o3[DM];
    __shared__ float red[2];

    for (int i = tid; i < KTOK * DM; i += 256)
        u[i / DM][i % DM] = tokens[b * KTOK * DM + i];
    __syncthreads();

    // in-projection: xz = u @ W_in  [4 x 608]
    for (int i = tid; i < KTOK * 2 * DINNER; i += 256) {
        int t = i / (2 * DINNER), j = i % (2 * DINNER);
        float s = 0.f;
        for (int c = 0; c < DM; ++c) s += u[t][c] * W_in[c * (2 * DINNER) + j];
        xz[t][j] = s;
    }
    __syncthreads();

    // causal depthwise conv + silu
    for (int i = tid; i < KTOK * DINNER; i += 256) {
        int t = i / DINNER, d = i % DINNER;
        float s = conv_b[d];
#pragma unroll
        for (int j = 0; j < DCONV; ++j) {
            int tt = t - (DCONV - 1) + j;
            if (tt >= 0) s += xz[tt][d] * conv_w[d * DCONV + j];
        }
        xc[t][d] = s / (1.0f + expf(-s));   // silu
    }
    __syncthreads();

    // x-projection: proj = xc @ W_xproj  [4 x 266]
    for (int i = tid; i < KTOK * (DTRANK + 2 * DSTATE); i += 256) {
        int t = i / (DTRANK + 2 * DSTATE), p = i % (DTRANK + 2 * DSTATE);
        float s = 0.f;
        for (int d = 0; d < DINNER; ++d)
            s += xc[t][d] * W_xproj[d * (DTRANK + 2 * DSTATE) + p];
        proj[t][p] = s;
    }
    __syncthreads();

    // dt = softplus(dt_in @ W_dtproj + b)
    for (int i = tid; i < KTOK * DINNER; i += 256) {
        int t = i / DINNER, d = i % DINNER;
        float s = b_dtproj[d];
#pragma unroll
        for (int r = 0; r < DTRANK; ++r) s += proj[t][r] * W_dtproj[r * DINNER + d];
        dtv[t][d] = (s > 20.0f) ? s : log1pf(expf(s));
    }
    for (int i = tid; i < DINNER; i += 256) y3[i] = 0.f;
    __syncthreads();

    // selective scan; accumulate only last-token output
    for (int p = tid; p < DINNER * DSTATE; p += 256) {
        int d = p >> 7, s = p & 127;
        float a = -expf(A_log[d * DSTATE + s]);
        float h = 0.f;
#pragma unroll
        for (int t = 0; t < KTOK; ++t) {
            float dtd = dtv[t][d];
            h = expf(dtd * a) * h + dtd * proj[t][DTRANK + s] * xc[t][d];
        }
        atomicAdd(&y3[d], h * proj[KTOK - 1][DTRANK + DSTATE + s]);
    }
    __syncthreads();

    for (int i = tid; i < DINNER; i += 256) {
        float zz = xz[KTOK - 1][DINNER + i];
        val[i] = (y3[i] + D_m[i] * xc[KTOK - 1][i]) * (zz / (1.0f + expf(-zz)));
    }
    __syncthreads();

    // out-projection (last token only)
    for (int m = tid; m < DM; m += 256) {
        float s = 0.f;
        for (int d = 0; d < DINNER; ++d) s += val[d] * W_outp[d * DM + m];
        o3[m] = s;
    }
    __syncthreads();

    if (tid == 0) {
        float mu = 0.f;
        for (int m = 0; m < DM; ++m) mu += o3[m];
        mu /= (float)DM;
        float var = 0.f;
        for (int m = 0; m < DM; ++m) { float d = o3[m] - mu; var += d * d; }
        var /= (float)DM;
        red[0] = mu;
        red[1] = rsqrtf(var + 1e-5f);
    }
    __syncthreads();
    for (int m = tid; m < DM; m += 256)
        lastv[b * DM + m] = (o3[m] - red[0]) * red[1] * g_lnm[m] + b_lnm[m];
}

// global_add_pool: g[b, c] = sum_n h3[b, n, c]
__global__ __launch_bounds__(128) void pool_kernel(const float* __restrict__ h3,
                                                   float* __restrict__ g)
{
    int b = blockIdx.x, c = threadIdx.x;
    float s = 0.f;
    for (int n = 0; n < NNODE; ++n) s += h3[(b * NNODE + n) * 128 + c];
    g[b * 128 + c] = s;
}

// =====================================================================
extern "C" void kernel_launch(void* const* d_in, const int* in_sizes, int n_in,
                              void* d_out, int out_size, void* d_ws, size_t ws_size,
                              hipStream_t stream)
{
    const float* x       = (const float*)d_in[0];
    const float* lpe     = (const float*)d_in[1];
    const float* rwse    = (const float*)d_in[2];
    const float* W_enc1  = (const float*)d_in[3];
    const float* b_enc1  = (const float*)d_in[4];
    const float* g_enc   = (const float*)d_in[5];
    const float* be_enc  = (const float*)d_in[6];
    const float* W_enc2  = (const float*)d_in[7];
    const float* b_enc2  = (const float*)d_in[8];
    const float* eps_gin = (const float*)d_in[9];
    const float* Wg1     = (const float*)d_in[10];
    const float* bg1     = (const float*)d_in[11];
    const float* g_gin   = (const float*)d_in[12];
    const float* be_gin  = (const float*)d_in[13];
    const float* Wg2     = (const float*)d_in[14];
    const float* bg2     = (const float*)d_in[15];
    const float* W_theta = (const float*)d_in[16];
    const float* W_in    = (const float*)d_in[17];
    const float* conv_w  = (const float*)d_in[18];
    const float* conv_b  = (const float*)d_in[19];
    const float* W_xproj = (const float*)d_in[20];
    const float* W_dtproj= (const float*)d_in[21];
    const float* b_dtproj= (const float*)d_in[22];
    const float* A_log   = (const float*)d_in[23];
    const float* D_m     = (const float*)d_in[24];
    const float* W_outp  = (const float*)d_in[25];
    const float* g_lnm   = (const float*)d_in[26];
    const float* b_lnm   = (const float*)d_in[27];
    const float* W_merge = (const float*)d_in[28];
    const float* b_merge = (const float*)d_in[29];
    const float* W_og    = (const float*)d_in[30];
    const float* b_og    = (const float*)d_in[31];
    const float* W_d1    = (const float*)d_in[32];
    const float* b_d1    = (const float*)d_in[33];
    const float* g_dec   = (const float*)d_in[34];
    const float* be_dec  = (const float*)d_in[35];
    const float* W_d2    = (const float*)d_in[36];
    const float* b_d2    = (const float*)d_in[37];
    const int*   eidx    = (const int*)d_in[38];
    const int*   batch   = (const int*)d_in[39];
    const int* src = eidx;
    const int* dst = eidx + EDGES;

    const int NH = NTOT * HID;   // 4,194,304 floats
    float* ws     = (float*)d_ws;
    float* bufA   = ws;              // h1 -> hm -> h3 (in place)
    float* bufB   = ws + (size_t)NH; // agg -> agg2
    float* h2     = ws + 2 * (size_t)NH;
    float* alpha  = ws + 3 * (size_t)NH;          // NTOT*4
    float* tokens = alpha + (size_t)NTOT * 4;     // 64*4*152
    float* lastv  = tokens + (size_t)BSZ * KTOK * DM;
    float* pooled = lastv + (size_t)BSZ * DM;     // 64*128

    // 1) encoder MLP: h1 = mlp2(x)
    mlp2_wmma_kernel<128><<<NTOT / 64, 128, 0, stream>>>(
        x, nullptr, nullptr, W_enc1, b_enc1, g_enc, be_enc, W_enc2, b_enc2, bufA);

    // 2) agg = segment_sum(h1[src] -> dst)
    zero_kernel<<<(NH + 255) / 256, 256, 0, stream>>>(bufB, NH);
    segsum_kernel<<<(EDGES * 32) / 256, 256, 0, stream>>>(bufA, src, dst, bufB);

    // 3) GIN MLP: h2 = mlp2((1+eps)*h1 + agg)
    mlp2_wmma_kernel<128><<<NTOT / 64, 128, 0, stream>>>(
        bufA, bufB, eps_gin, Wg1, bg1, g_gin, be_gin, Wg2, bg2, h2);

    // 4) alpha = hp @ W_theta
    alpha_kernel<<<NTOT / 256, 256, 0, stream>>>(h2, lpe, rwse, W_theta, alpha);

    // 5) token select (argmax per slot, /N)
    tokens_kernel<<<BSZ, 256, 0, stream>>>(alpha, h2, lpe, rwse, tokens);

    // 6) mamba block -> LN -> last token
    mamba_kernel<<<BSZ, 256, 0, stream>>>(
        tokens, W_in, conv_w, conv_b, W_xproj, W_dtproj, b_dtproj,
        A_log, D_m, W_outp, g_lnm, b_lnm, lastv);

    // 7) merge GEMM: hm = [h2 | last[batch]] @ W_merge + b   (reuses bufA)
    merge_wmma_kernel<<<NTOT / 32, 64, 0, stream>>>(
        h2, lastv, batch, W_merge, b_merge, bufA);

    // 8) agg2 = segment_sum(hm[src] -> dst)
    zero_kernel<<<(NH + 255) / 256, 256, 0, stream>>>(bufB, NH);
    segsum_kernel<<<(EDGES * 32) / 256, 256, 0, stream>>>(bufA, src, dst, bufB);

    // 9) output GIN: h3 = (hm + agg2) @ W_og + b_og   (in-place per-WG safe)
    gemm128_wmma_kernel<<<NTOT / 64, 128, 0, stream>>>(bufA, bufB, W_og, b_og, bufA);

    // 10) global add pool
    pool_kernel<<<BSZ, 128, 0, stream>>>(bufA, pooled);

    // 11) decoder: out = mlp2(pooled) -> [64, 32]
    mlp2_wmma_kernel<32><<<1, 128, 0, stream>>>(
        pooled, nullptr, nullptr, W_d1, b_d1, g_dec, be_dec, W_d2, b_d2,
        (float*)d_out);
}